// PreProcesser_43593918055039
// MI455X (gfx1250) — compile-verified
//
#include <hip/hip_runtime.h>

#define ZDIM 4
#define NRES 512
#define NAT  14
#define NVOX 288          // 6*8*6
#define TOPK 30
#define X_LBL 20

typedef __attribute__((ext_vector_type(2))) float v2f;
typedef __attribute__((ext_vector_type(8))) float v8f;

// ---------------------------------------------------------------------------
// Kernel 1: backbone (Cb) + local frames. One thread per (z,n) residue.
// ---------------------------------------------------------------------------
__global__ __launch_bounds__(256) void backbone_frames_kernel(
    const float* __restrict__ C, float* __restrict__ cb_out,
    float* __restrict__ frames)
{
  int zn = blockIdx.x * blockDim.x + threadIdx.x;
  if (zn >= ZDIM * NRES) return;
  const float* p = C + (size_t)zn * NAT * 3;
  float nx  = p[0], ny  = p[1], nz  = p[2];
  float cax = p[3], cay = p[4], caz = p[5];
  float cx  = p[6], cy  = p[7], cz  = p[8];
  float b1x = cax - nx,  b1y = cay - ny,  b1z = caz - nz;
  float b2x = cx - cax,  b2y = cy - cay,  b2z = cz - caz;
  float b3x = b1y * b2z - b1z * b2y;
  float b3y = b1z * b2x - b1x * b2z;
  float b3z = b1x * b2y - b1y * b2x;
  const float kb2 = -0.58273431f, kb1 = 0.56802827f, kb3 = -0.54067466f;
  float cbx = cax + kb2 * b2x + kb1 * b1x + kb3 * b3x;
  float cby = cay + kb2 * b2y + kb1 * b1y + kb3 * b3y;
  float cbz = caz + kb2 * b2z + kb1 * b1z + kb3 * b3z;

  float* o = cb_out + (size_t)zn * 4 * 3;
  o[0] = nx;  o[1]  = ny;  o[2]  = nz;
  o[3] = cax; o[4]  = cay; o[5]  = caz;
  o[6] = cx;  o[7]  = cy;  o[8]  = cz;
  o[9] = cbx; o[10] = cby; o[11] = cbz;

  // frames: rows = [x_u, y_u, z_u]
  float yx = cbx - cax, yy = cby - cay, yz = cbz - caz;
  float ynorm = fmaxf(sqrtf(yx * yx + yy * yy + yz * yz), 1e-6f);
  float yux = yx / ynorm, yuy = yy / ynorm, yuz = yz / ynorm;
  float xrx = cx - nx, xry = cy - ny, xrz = cz - nz;
  float xp = xrx * yux + xry * yuy + xrz * yuz;
  float xx = xrx - xp * yux, xy = xry - xp * yuy, xz = xrz - xp * yuz;
  float xnorm = fmaxf(sqrtf(xx * xx + xy * xy + xz * xz), 1e-6f);
  float xux = xx / xnorm, xuy = xy / xnorm, xuz = xz / xnorm;
  float zux = xuy * yuz - xuz * yuy;
  float zuy = xuz * yux - xux * yuz;
  float zuz = xux * yuy - xuy * yux;
  float* f = frames + (size_t)zn * 9;
  f[0] = xux; f[1] = xuy; f[2] = xuz;
  f[3] = yux; f[4] = yuy; f[5] = yuz;
  f[6] = zux; f[7] = zuy; f[8] = zuz;
}

// ---------------------------------------------------------------------------
// Kernel 2: top-30 nearest neighbors per residue. One wave32 per residue.
// Distances live in registers (16 per lane); 30 rounds of packed 64-bit
// (dist,index) min-reduction reproduce jax.lax.top_k's ascending-index
// tie-break exactly.
// ---------------------------------------------------------------------------
__global__ __launch_bounds__(32) void neighbors_kernel(
    const float* __restrict__ C, const int* __restrict__ kp,
    int* __restrict__ nbrs_i, int* __restrict__ mask_i,
    float* __restrict__ nbrs_f, float* __restrict__ mask_f)
{
  int zn = blockIdx.x;
  int z = zn >> 9, i = zn & (NRES - 1);
  int lane = threadIdx.x;
  const float* base = C + (size_t)z * NRES * NAT * 3;
  const float* ps = base + ((size_t)i * NAT + 1) * 3;   // Ca of self
  float sx = ps[0], sy = ps[1], sz = ps[2];
  int kp_i = kp[zn];

  float dloc[NRES / 32];
#pragma unroll
  for (int t = 0; t < NRES / 32; ++t) {
    int j = lane + t * 32;
    const float* pj = base + ((size_t)j * NAT + 1) * 3;
    float dx = pj[0] - sx, dy = pj[1] - sy, dz = pj[2] - sz;
    float d = sqrtf(dx * dx + dy * dy + dz * dz);
    if (d == 0.0f || !kp_i) d = __builtin_inff();
    dloc[t] = d;
  }

  for (int k = 0; k < TOPK; ++k) {
    unsigned long long best = ~0ull;
#pragma unroll
    for (int t = 0; t < NRES / 32; ++t) {
      unsigned long long key =
          ((unsigned long long)__float_as_uint(dloc[t]) << 32) |
          (unsigned)(lane + t * 32);
      best = key < best ? key : best;
    }
#pragma unroll
    for (int off = 16; off > 0; off >>= 1) {
      unsigned long long other = __shfl_down(best, off, 32);
      best = other < best ? other : best;
    }
    best = __shfl(best, 0, 32);
    int idx = (int)(best & 0xffffffffu);
    if (lane == (idx & 31)) dloc[idx >> 5] = __builtin_inff();
    if (lane == 0) {
      int kp_j = kp[z * NRES + idx];
      int m = (!kp_i && !kp_j) ? 1 : 0;   // vals!=0 always true (masked d>0)
      int nb = m ? idx : i;
      int o = zn * TOPK + k;
      nbrs_i[o] = nb;
      mask_i[o] = m;
      nbrs_f[o] = (float)nb;
      mask_f[o] = (float)m;
    }
  }
}

// ---------------------------------------------------------------------------
// Kernel 3: electric fields. One block of 288 threads (9 wave32s) per residue;
// thread v owns voxel v. Voxel positions computed via V_WMMA_F32_16X16X4_F32:
//   D = vox(16x4,K=3 used) @ frame(4x16,3 cols used) + C(origin broadcast).
// ---------------------------------------------------------------------------
__global__ __launch_bounds__(NVOX) void fields_kernel(
    const float* __restrict__ C, const int* __restrict__ L,
    const float* __restrict__ atom_mask, const float* __restrict__ charges,
    const float* __restrict__ frames, const float* __restrict__ cb_out,
    const int* __restrict__ nbrs_i, const int* __restrict__ mask_i,
    float* __restrict__ fields_out)
{
  __shared__ float sF[12];            // 9 frame entries + 3 origin
  __shared__ float sVox[NVOX * 3];
  __shared__ float sAtom[NAT * 4];    // x,y,z,q per atom
  __shared__ float sRed[NVOX];
  __shared__ int   sNbr[TOPK];
  __shared__ float sStat[2];

  int zn = blockIdx.x;
  int z = zn >> 9, nres = zn & (NRES - 1);
  int tid = threadIdx.x;

  if (tid < 9) sF[tid] = frames[(size_t)zn * 9 + tid];
  else if (tid < 12) sF[tid] = cb_out[((size_t)zn * 4 + 3) * 3 + (tid - 9)];
  if (tid < TOPK) {
    int m = mask_i[zn * TOPK + tid];
    sNbr[tid] = m ? nbrs_i[zn * TOPK + tid] : -1;   // -1 => zero contribution
  }
  __syncthreads();

  // ---- WMMA voxel transform: 2 tiles of 16 voxels per wave ----
  {
    int wave = tid >> 5, lane = tid & 31;
    int half = lane >> 4, ln = lane & 15;
    // B (4x16): rows K=0,1 in lanes 0-15; rows K=2,3 in lanes 16-31
    float bx, by;
    if (half == 0) { bx = (ln < 3) ? sF[ln]     : 0.f;
                     by = (ln < 3) ? sF[3 + ln] : 0.f; }
    else           { bx = (ln < 3) ? sF[6 + ln] : 0.f;
                     by = 0.f; }
    v2f B = {bx, by};
    float cv = (ln < 3) ? sF[9 + ln] : 0.f;          // origin per column
    v8f Cm = {cv, cv, cv, cv, cv, cv, cv, cv};
#pragma unroll
    for (int t = 0; t < 2; ++t) {
      int rb = wave * 32 + t * 16;                   // tile base voxel row
      int row = rb + ln;
      float vx = (float)(row / 48)      - 3.0f;      // ix - 6/2
      float vy = (float)((row / 6) % 8) - 2.0f;      // iy - 8/4
      float vz = (float)(row % 6)       - 3.0f;      // iz - 6/2
      // A (16x4): K=0,1 in lanes 0-15; K=2,3 in lanes 16-31
      v2f A;
      A.x = half ? vz : vx;
      A.y = half ? 0.f : vy;
      v8f D = __builtin_amdgcn_wmma_f32_16x16x4_f32(
          false, A, false, B, (short)0, Cm, false, false);
      if (ln < 3) {                                  // columns 0..2 are x,y,z
        int r0 = rb + half * 8;
#pragma unroll
        for (int r = 0; r < 8; ++r) sVox[(r0 + r) * 3 + ln] = D[r];
      }
    }
  }
  __syncthreads();

  float px = sVox[tid * 3 + 0];
  float py = sVox[tid * 3 + 1];
  float pz = sVox[tid * 3 + 2];
  float accx = 0.f, accy = 0.f, accz = 0.f;

  for (int slot = 0; slot <= TOPK; ++slot) {
    int idx = (slot < TOPK) ? sNbr[slot] : nres;     // slot 30 = self, mask=1
    if (idx < 0) continue;                           // uniform across block
    __syncthreads();
    if (tid < NAT) {
      size_t ri = (size_t)z * NRES + idx;
      const float* cp = C + (ri * NAT + tid) * 3;
      int lv = L[ri]; if (lv == -1) lv = X_LBL;
      float q = charges[lv * NAT + tid] * atom_mask[ri * NAT + tid];
      sAtom[tid * 4 + 0] = cp[0];
      sAtom[tid * 4 + 1] = cp[1];
      sAtom[tid * 4 + 2] = cp[2];
      sAtom[tid * 4 + 3] = q;
    }
    if (tid == 0 && slot < TOPK) {                   // gfx1250 global_prefetch
      int nidx = (slot + 1 < TOPK) ? sNbr[slot + 1] : nres;
      if (nidx >= 0)
        __builtin_prefetch(C + (((size_t)z * NRES + nidx) * NAT) * 3, 0, 1);
    }
    __syncthreads();
#pragma unroll
    for (int a = 0; a < NAT; ++a) {
      float dx = px - sAtom[a * 4 + 0];
      float dy = py - sAtom[a * 4 + 1];
      float dz = pz - sAtom[a * 4 + 2];
      float r2 = dx * dx + dy * dy + dz * dz;
      float w = sAtom[a * 4 + 3] / fmaxf(r2, 4.0f);  // clip(|dv|,2)^2=max(r2,4)
      accx += w * dx; accy += w * dy; accz += w * dz;
    }
  }

  // ---- normalization: mean + ddof=1 std over the 864 values ----
  sRed[tid] = accx + accy + accz;
  __syncthreads();
  if (tid < 144) sRed[tid] += sRed[tid + 144]; __syncthreads();
  if (tid < 72)  sRed[tid] += sRed[tid + 72];  __syncthreads();
  if (tid < 36)  sRed[tid] += sRed[tid + 36];  __syncthreads();
  if (tid < 18)  sRed[tid] += sRed[tid + 18];  __syncthreads();
  if (tid < 9)   sRed[tid] += sRed[tid + 9];   __syncthreads();
  if (tid == 0) {
    float t = 0.f;
    for (int q = 0; q < 9; ++q) t += sRed[q];
    sStat[0] = t * (1.0f / (NVOX * 3.0f));
  }
  __syncthreads();
  float mean = sStat[0];
  float c0 = accx - mean, c1 = accy - mean, c2 = accz - mean;
  sRed[tid] = c0 * c0 + c1 * c1 + c2 * c2;
  __syncthreads();
  if (tid < 144) sRed[tid] += sRed[tid + 144]; __syncthreads();
  if (tid < 72)  sRed[tid] += sRed[tid + 72];  __syncthreads();
  if (tid < 36)  sRed[tid] += sRed[tid + 36];  __syncthreads();
  if (tid < 18)  sRed[tid] += sRed[tid + 18];  __syncthreads();
  if (tid < 9)   sRed[tid] += sRed[tid + 9];   __syncthreads();
  if (tid == 0) {
    float t = 0.f;
    for (int q = 0; q < 9; ++q) t += sRed[q];
    float sd = sqrtf(t / (NVOX * 3.0f - 1.0f));      // ddof = 1
    sStat[1] = (sd == 0.0f) ? 1.0f : sd;
  }
  __syncthreads();
  float sd = sStat[1];
  float* op = fields_out + ((size_t)zn * NVOX + tid) * 3;
  op[0] = c0 / sd;
  op[1] = c1 / sd;
  op[2] = c2 / sd;
}

// ---------------------------------------------------------------------------
extern "C" void kernel_launch(void* const* d_in, const int* in_sizes, int n_in,
                              void* d_out, int out_size, void* d_ws, size_t ws_size,
                              hipStream_t stream) {
  (void)in_sizes; (void)n_in; (void)out_size; (void)ws_size;
  const float* C         = (const float*)d_in[0];
  const int*   L         = (const int*)d_in[1];
  const float* atom_mask = (const float*)d_in[2];
  const int*   kp        = (const int*)d_in[3];
  const float* charges   = (const float*)d_in[4];

  const int ZN = ZDIM * NRES;                 // 2048 residues
  float* out        = (float*)d_out;
  float* cb_out     = out;                                 // ZN*4*3
  float* fields_out = out + (size_t)ZN * 4 * 3;            // ZN*288*3
  float* nbrs_f     = fields_out + (size_t)ZN * NVOX * 3;  // ZN*30
  float* mask_f     = nbrs_f + (size_t)ZN * TOPK;          // ZN*30

  char* wsb = (char*)d_ws;
  float* frames = (float*)wsb;                                  // ZN*9 f32
  int* nbrs_i = (int*)(wsb + (size_t)ZN * 9 * sizeof(float));   // ZN*30 i32
  int* mask_i = (int*)(wsb + (size_t)ZN * 9 * sizeof(float)
                           + (size_t)ZN * TOPK * sizeof(int));  // ZN*30 i32

  backbone_frames_kernel<<<(ZN + 255) / 256, 256, 0, stream>>>(C, cb_out, frames);
  neighbors_kernel<<<ZN, 32, 0, stream>>>(C, kp, nbrs_i, mask_i, nbrs_f, mask_f);
  fields_kernel<<<ZN, NVOX, 0, stream>>>(C, L, atom_mask, charges, frames,
                                         cb_out, nbrs_i, mask_i, fields_out);
}